// SpanCAT_39599598469527
// MI455X (gfx1250) — compile-verified
//
#include <hip/hip_runtime.h>
#include <hip/hip_bf16.h>

typedef float v2f __attribute__((ext_vector_type(2)));
typedef float v8f __attribute__((ext_vector_type(8)));

#define HIDDEN 768
#define FANIN  896   // HIDDEN + Q_DIM
#define QDIM   128
#define MAXW   12
#define BM 128
#define BN 128
#define BK 32
#define LDA 33       // padded LDS stride for A to avoid bank conflicts

// ---------------------------------------------------------------------------
// Kernel 1: qb[w][o] = b[o] + sum_q q[w,q] * W[o, 768+q]   (12 x 768, tiny)
// ---------------------------------------------------------------------------
__global__ __launch_bounds__(256) void spancat_qb_kernel(
    const float* __restrict__ q,   // (12,128) flat
    const float* __restrict__ W,   // (768,896)
    const float* __restrict__ b,   // (768)
    float* __restrict__ qb)        // (12,768)
{
    int idx = blockIdx.x * blockDim.x + threadIdx.x;
    if (idx >= MAXW * HIDDEN) return;
    int w = idx / HIDDEN;
    int o = idx - w * HIDDEN;
    const float4* qv = (const float4*)(q + w * QDIM);
    const float4* wv = (const float4*)(W + (size_t)o * FANIN + HIDDEN);
    float acc = b[o];
#pragma unroll
    for (int i = 0; i < QDIM / 4; ++i) {
        float4 a = qv[i], c = wv[i];
        acc += a.x * c.x + a.y * c.y + a.z * c.z + a.w * c.w;
    }
    qb[idx] = acc;
}

// ---------------------------------------------------------------------------
// Kernel 2: fused GEMM (h @ W_h^T) + 12-way broadcast add(qb) + relu
// out[(m*12+w)*768 + o] = relu(hpart[m,o] + qb[w,o]),  m in [0,8192)
// ---------------------------------------------------------------------------
__global__ __launch_bounds__(256) void spancat_gemm_kernel(
    const float* __restrict__ h,    // (8192,768)
    const float* __restrict__ W,    // (768,896) -> W_h = [:, :768]
    const float* __restrict__ qb,   // (12,768)
    float* __restrict__ out)        // (8192,12,768)
{
    __shared__ float As[BM * LDA];  // [m][k] padded
    __shared__ float Bs[BK * BN];   // [k][n]

    const int tid   = threadIdx.x;
    const int lane  = tid & 31;
    const int wave  = tid >> 5;
    const int waveM = wave & 3;     // 4 waves along M  -> 32 rows each
    const int waveN = wave >> 2;    // 2 waves along N  -> 64 cols each
    const int m_base = blockIdx.x * BM;
    const int n_base = blockIdx.y * BN;

    const int l16   = lane & 15;
    const int khalf = (lane >> 4) << 1;  // lanes 0-15: K=+0,+1 ; lanes 16-31: K=+2,+3

    v8f acc[2][4];
#pragma unroll
    for (int i = 0; i < 2; ++i)
#pragma unroll
        for (int j = 0; j < 4; ++j) acc[i][j] = (v8f)0.0f;

    for (int kb = 0; kb < HIDDEN; kb += BK) {
        // ---- stage A tile: 128 x 32, float4-coalesced ----
#pragma unroll
        for (int i = 0; i < 4; ++i) {
            int f   = tid + i * 256;     // float4 id, 0..1023
            int row = f >> 3;            // 0..127
            int kq  = (f & 7) << 2;      // 0,4,...,28
            const float4 v =
                *(const float4*)(h + (size_t)(m_base + row) * HIDDEN + kb + kq);
            float* dst = &As[row * LDA + kq];
            dst[0] = v.x; dst[1] = v.y; dst[2] = v.z; dst[3] = v.w;
        }
        // ---- stage B tile transposed: Bs[k][n] = W[(n_base+n)*896 + kb + k] ----
        {
            int n    = tid & 127;
            int half = tid >> 7;         // 0 or 1 -> k range 0..15 / 16..31
            const float* src = W + (size_t)(n_base + n) * FANIN + kb + half * 16;
#pragma unroll
            for (int j = 0; j < 4; ++j) {
                float4 v = *(const float4*)(src + j * 4);
                int k = half * 16 + j * 4;
                Bs[(k + 0) * BN + n] = v.x;
                Bs[(k + 1) * BN + n] = v.y;
                Bs[(k + 2) * BN + n] = v.z;
                Bs[(k + 3) * BN + n] = v.w;
            }
        }
        __syncthreads();

        // ---- 8 k-steps of V_WMMA_F32_16X16X4_F32 per stage ----
#pragma unroll
        for (int k = 0; k < BK; k += 4) {
            v2f afrag[2];
#pragma unroll
            for (int mt = 0; mt < 2; ++mt) {
                const float* ap = &As[(waveM * 32 + mt * 16 + l16) * LDA + k + khalf];
                afrag[mt].x = ap[0];     // K = k+khalf
                afrag[mt].y = ap[1];     // K = k+khalf+1
            }
            v2f bfrag[4];
#pragma unroll
            for (int nt = 0; nt < 4; ++nt) {
                const float* bp = &Bs[(k + khalf) * BN + waveN * 64 + nt * 16 + l16];
                bfrag[nt].x = bp[0];
                bfrag[nt].y = bp[BN];
            }
#pragma unroll
            for (int mt = 0; mt < 2; ++mt)
#pragma unroll
                for (int nt = 0; nt < 4; ++nt)
                    acc[mt][nt] = __builtin_amdgcn_wmma_f32_16x16x4_f32(
                        false, afrag[mt], false, bfrag[nt],
                        (short)0, acc[mt][nt], false, false);
        }
        __syncthreads();
    }

    // ---- epilogue: 12-way broadcast + relu, streaming NT stores ----
    // C layout (16x16 f32): VGPR r, lanes 0-15 -> M=r, lanes 16-31 -> M=r+8, N=lane%16
    const int m_wave = m_base + waveM * 32;
    const int m_half = (lane >> 4) << 3;   // +0 or +8
#pragma unroll
    for (int nt = 0; nt < 4; ++nt) {
        const int o = n_base + waveN * 64 + nt * 16 + l16;
        for (int w = 0; w < MAXW; ++w) {
            const float qv = qb[w * HIDDEN + o];
#pragma unroll
            for (int mt = 0; mt < 2; ++mt) {
                const int mrow = m_wave + mt * 16 + m_half;
#pragma unroll
                for (int r = 0; r < 8; ++r) {
                    float val = acc[mt][nt][r] + qv;
                    val = val > 0.0f ? val : 0.0f;
                    __builtin_nontemporal_store(
                        val, out + ((size_t)(mrow + r) * MAXW + w) * HIDDEN + o);
                }
            }
        }
    }
}

// ---------------------------------------------------------------------------
extern "C" void kernel_launch(void* const* d_in, const int* in_sizes, int n_in,
                              void* d_out, int out_size, void* d_ws, size_t ws_size,
                              hipStream_t stream) {
    const float* h  = (const float*)d_in[0];   // (8,1024,768)
    const float* q  = (const float*)d_in[1];   // (128,12) flat == (12,128) reshape
    const float* W  = (const float*)d_in[2];   // (768,896)
    const float* b  = (const float*)d_in[3];   // (768)
    float* out = (float*)d_out;                // (8,1024,12,768)
    float* qb  = (float*)d_ws;                 // 12*768 floats scratch

    // 1) q_part + bias -> qb
    {
        int total = MAXW * HIDDEN;             // 9216
        spancat_qb_kernel<<<(total + 255) / 256, 256, 0, stream>>>(q, W, b, qb);
    }
    // 2) fused WMMA GEMM + broadcast epilogue
    {
        dim3 grid(8192 / BM, HIDDEN / BN);     // (64, 6)
        spancat_gemm_kernel<<<grid, 256, 0, stream>>>(h, W, qb, out);
    }
}